// MultiBoxLoss_71708773974165
// MI455X (gfx1250) — compile-verified
//
#include <hip/hip_runtime.h>
#include <stdint.h>
#include <math.h>

#define B_ 32
#define P_ 8732
#define C_ 81
#define M_ 20
#define THRESH 0.5f
#define SORT_N 16384   // next pow2 >= P_

typedef __attribute__((ext_vector_type(2))) float v2f;
typedef __attribute__((ext_vector_type(8))) float v8f;

// ---------------------------------------------------------------------------
// Exact fp32 wave(32) sum via V_WMMA_F32_16X16X4_F32 against a ones B-matrix.
// A layout (32-bit, 16x4): lane L<16 -> A[M=L][K=0] = v, lane L>=16 -> A[M=L-16][K=2] = v,
// second A VGPR (K=1/K=3) = 0.  With B = ones, D[m][n] = v[m] + v[m+16] for all n.
// D VGPR j on lanes 0-15 holds M=j, lanes 16-31 hold M=8+j, so summing the 8 D
// registers gives the low-half / high-half partial on each half-wave; one
// shfl_xor(16) completes the full 32-lane sum (all lanes get the total).
// ---------------------------------------------------------------------------
__device__ __forceinline__ float wave_sum32(float v) {
#if __has_builtin(__builtin_amdgcn_wmma_f32_16x16x4_f32)
    v2f a;    a[0] = v;    a[1] = 0.0f;
    v2f ones; ones[0] = 1.0f; ones[1] = 1.0f;
    v8f c = {};
    v8f d = __builtin_amdgcn_wmma_f32_16x16x4_f32(false, a, false, ones,
                                                  (short)0, c, false, false);
    float s = ((d[0] + d[1]) + (d[2] + d[3])) + ((d[4] + d[5]) + (d[6] + d[7]));
    s += __shfl_xor(s, 16, 32);
    return s;
#else
    for (int off = 16; off > 0; off >>= 1) v += __shfl_xor(v, off, 32);
    return v;
#endif
}

__device__ __forceinline__ float smooth_l1(float d) {
    float ad = fabsf(d);
    return (ad < 1.0f) ? 0.5f * d * d : ad - 0.5f;
}

// ---------------------------------------------------------------------------
// Kernel 0: zero per-launch accumulators (graph-replay safe).
// scal[0]=loc_sum  scal[1]=conf_pos  scal[2]=conf_neg
// ---------------------------------------------------------------------------
__global__ void k_init(unsigned long long* best, int* npos, float* scal) {
    int t = threadIdx.x;
    for (int i = t; i < B_ * M_; i += 256) best[i] = 0ull;
    if (t < B_) npos[t] = 0;
    if (t < 4)  scal[t] = 0.0f;
}

// ---------------------------------------------------------------------------
// Kernel 1: IoU matching. One image per blockIdx.y, priors strided over x.
// Per prior: max/argmax over the 20 boxes. Per object: packed-max
// (iou_bits<<32 | ~p) via LDS then one global u64 atomic per block.
// ---------------------------------------------------------------------------
__global__ void k_match(const float4* __restrict__ priors,
                        const float4* __restrict__ boxes,
                        float* __restrict__ ov, int* __restrict__ obj,
                        unsigned long long* __restrict__ best) {
    __shared__ float4 sbox[M_];
    __shared__ float  sarea[M_];
    __shared__ unsigned long long sbest[M_];
    int b = blockIdx.y, t = threadIdx.x;
    int p = blockIdx.x * 256 + t;
    if (t < M_) {
        float4 bx = boxes[b * M_ + t];
        sbox[t]  = bx;
        sarea[t] = (bx.z - bx.x) * (bx.w - bx.y);
        sbest[t] = 0ull;
    }
    __syncthreads();
    if (p < P_) {
        float4 pc = priors[p];
        float px1 = pc.x - pc.z * 0.5f, py1 = pc.y - pc.w * 0.5f;
        float px2 = pc.x + pc.z * 0.5f, py2 = pc.y + pc.w * 0.5f;
        float pa  = pc.z * pc.w;
        float bestov = -1.0f; int bestm = 0;
#pragma unroll
        for (int m = 0; m < M_; ++m) {
            float4 bx = sbox[m];
            float w = fmaxf(fminf(px2, bx.z) - fmaxf(px1, bx.x), 0.0f);
            float h = fmaxf(fminf(py2, bx.w) - fmaxf(py1, bx.y), 0.0f);
            float inter = w * h;
            float iou = inter / (pa + sarea[m] - inter);
            if (iou > bestov) { bestov = iou; bestm = m; }
            unsigned long long key =
                ((unsigned long long)__float_as_uint(iou) << 32) | (unsigned)(~p);
            atomicMax(&sbest[m], key);
        }
        ov[b * P_ + p]  = bestov;
        obj[b * P_ + p] = bestm;
    }
    __syncthreads();
    if (t < M_) atomicMax(&best[b * M_ + t], sbest[t]);
}

// Kernel 2: force-assign each object to its best prior.
__global__ void k_force(const unsigned long long* __restrict__ best,
                        float* __restrict__ ov, int* __restrict__ obj) {
    int i = blockIdx.x * 64 + threadIdx.x;
    if (i >= B_ * M_) return;
    int b = i / M_, m = i % M_;
    unsigned p = ~(unsigned)(best[i] & 0xFFFFFFFFull);
    if (p < P_) { obj[b * P_ + p] = m; ov[b * P_ + p] = 1.0f; }
}

// ---------------------------------------------------------------------------
// Kernel 3: smooth-L1 over positives + n_pos counting.
// Thread-per-prior; WMMA wave-sum then 8-wave LDS combine, one atomic/block.
// ---------------------------------------------------------------------------
__global__ void k_loc(const float4* __restrict__ priors,
                      const float4* __restrict__ boxes,
                      const int* __restrict__ labels,
                      const float4* __restrict__ plocs,
                      const float* __restrict__ ov, const int* __restrict__ obj,
                      float* __restrict__ scal, int* __restrict__ npos) {
    __shared__ float wsum[8], wcnt[8];
    int b = blockIdx.y, t = threadIdx.x;
    int p = blockIdx.x * 256 + t;
    float sl1 = 0.0f, cnt = 0.0f;
    if (p < P_) {
        float v = ov[b * P_ + p];
        if (v >= THRESH) {            // labels are always >= 1, so pos == (ov >= 0.5)
            int o = obj[b * P_ + p];
            (void)labels;
            float4 bx = boxes[b * M_ + o];
            float4 pc = priors[p];
            float cx = (bx.x + bx.z) * 0.5f, cy = (bx.y + bx.w) * 0.5f;
            float w  = bx.z - bx.x,          h  = bx.w - bx.y;
            float g0 = (cx - pc.x) / (pc.z * 0.1f);
            float g1 = (cy - pc.y) / (pc.w * 0.1f);
            float g2 = logf(w / pc.z) * 5.0f;
            float g3 = logf(h / pc.w) * 5.0f;
            float4 pl = plocs[(size_t)b * P_ + p];
            sl1 = smooth_l1(pl.x - g0) + smooth_l1(pl.y - g1) +
                  smooth_l1(pl.z - g2) + smooth_l1(pl.w - g3);
            cnt = 1.0f;
        }
    }
    float ssum = wave_sum32(sl1);
    float scnt = wave_sum32(cnt);
    if ((t & 31) == 0) { wsum[t >> 5] = ssum; wcnt[t >> 5] = scnt; }
    __syncthreads();
    if (t == 0) {
        float a = 0.0f, c = 0.0f;
        for (int i = 0; i < 8; ++i) { a += wsum[i]; c += wcnt[i]; }
        if (a != 0.0f) atomicAdd(&scal[0], a);
        int ci = (int)(c + 0.5f);
        if (ci) atomicAdd(&npos[b], ci);
    }
}

// ---------------------------------------------------------------------------
// Kernel 4: per-prior cross entropy (the 90 MB streaming pass).
// Wave per prior, lanes over the 81 classes (coalesced 128B/wave lines),
// non-temporal loads (read-once data).
// ---------------------------------------------------------------------------
__global__ void k_ce(const float* __restrict__ scores,
                     const int* __restrict__ labels,
                     const float* __restrict__ ov, const int* __restrict__ obj,
                     float* __restrict__ ceneg, float* __restrict__ scal) {
    __shared__ float sPos;
    int b = blockIdx.y, t = threadIdx.x;
    int wid = t >> 5, lane = t & 31;
    int p = blockIdx.x * 8 + wid;
    if (t == 0) sPos = 0.0f;
    __syncthreads();
    if (p < P_) {
        float v = ov[b * P_ + p];
        int o   = obj[b * P_ + p];
        int lab = (v < THRESH) ? 0 : labels[b * M_ + o];
        const float* sc = scores + ((size_t)b * P_ + p) * C_;
        float x0 = __builtin_nontemporal_load(sc + lane);
        float x1 = __builtin_nontemporal_load(sc + lane + 32);
        bool  h2 = (lane + 64) < C_;
        float x2 = h2 ? __builtin_nontemporal_load(sc + lane + 64) : -3.4e38f;
        float mx = fmaxf(fmaxf(x0, x1), x2);
        for (int off = 16; off > 0; off >>= 1) mx = fmaxf(mx, __shfl_xor(mx, off, 32));
        float es = expf(x0 - mx) + expf(x1 - mx) + (h2 ? expf(x2 - mx) : 0.0f);
        float tg = (lane == lab ? x0 : 0.0f) + (lane + 32 == lab ? x1 : 0.0f) +
                   ((h2 && lane + 64 == lab) ? x2 : 0.0f);
        for (int off = 16; off > 0; off >>= 1) {
            es += __shfl_xor(es, off, 32);
            tg += __shfl_xor(tg, off, 32);
        }
        if (lane == 0) {
            float ce = mx + logf(es) - tg;
            bool pos = lab > 0;
            ceneg[b * P_ + p] = pos ? 0.0f : ce;
            if (pos) atomicAdd(&sPos, ce);
        }
    }
    __syncthreads();
    if (t == 0 && sPos != 0.0f) atomicAdd(&scal[1], sPos);
}

// ---------------------------------------------------------------------------
// Kernel 5: hard-negative mining. One workgroup per image; the entire 8732-
// element CE vector lives in 64 KB of LDS (CDNA5 WGP has 320 KB), staged with
// async global->LDS loads (ASYNCcnt path), bitonic-sorted descending, then
// top-3*n_pos summed.
// ---------------------------------------------------------------------------
__global__ void __launch_bounds__(1024)
k_sort(const float* __restrict__ ceneg, const int* __restrict__ npos,
       float* __restrict__ scal) {
    __shared__ float buf[SORT_N];
    int b = blockIdx.x, t = threadIdx.x;
    const float* src = ceneg + (size_t)b * P_;
#if __has_builtin(__builtin_amdgcn_global_load_async_to_lds_b32)
    typedef __attribute__((address_space(1))) int* gp1_t;
    typedef __attribute__((address_space(3))) int* lp3_t;
    float* srcnc = const_cast<float*>(src);            // builtin wants non-const
    for (int i = t; i < P_; i += 1024)
        __builtin_amdgcn_global_load_async_to_lds_b32((gp1_t)(srcnc + i),
                                                      (lp3_t)(&buf[i]), 0, 0);
    for (int i = P_ + t; i < SORT_N; i += 1024) buf[i] = -1.0f;
    asm volatile("s_wait_asynccnt 0" ::: "memory");
#else
    for (int i = t; i < SORT_N; i += 1024) buf[i] = (i < P_) ? src[i] : -1.0f;
#endif
    __syncthreads();
    for (unsigned k = 2; k <= SORT_N; k <<= 1) {
        for (unsigned j = k >> 1; j > 0; j >>= 1) {
            for (unsigned i = t; i < SORT_N; i += 1024) {
                unsigned ix = i ^ j;
                if (ix > i) {
                    float a = buf[i], c = buf[ix];
                    bool desc = ((i & k) == 0);
                    if (desc ? (a < c) : (a > c)) { buf[i] = c; buf[ix] = a; }
                }
            }
            __syncthreads();
        }
    }
    int K = 3 * npos[b];
    if (K > P_) K = P_;
    float s = 0.0f;
    for (int i = t; i < K; i += 1024) s += buf[i];
    __syncthreads();
    buf[t] = s;
    __syncthreads();
    for (int off = 512; off > 0; off >>= 1) {
        if (t < off) buf[t] += buf[t + off];
        __syncthreads();
    }
    if (t == 0 && buf[0] != 0.0f) atomicAdd(&scal[2], buf[0]);
}

// Kernel 6: finalize the two scalars.
__global__ void k_final(const float* __restrict__ scal,
                        const int* __restrict__ npos, float* __restrict__ out) {
    if (threadIdx.x == 0 && blockIdx.x == 0) {
        int tot = 0;
        for (int b = 0; b < B_; ++b) tot += npos[b];
        float nt = (float)tot;
        out[0] = (scal[2] + scal[1]) / nt;         // conf_loss
        out[1] = scal[0] / (nt * 4.0f);            // ALPHA * loc_loss (ALPHA = 1)
    }
}

extern "C" void kernel_launch(void* const* d_in, const int* in_sizes, int n_in,
                              void* d_out, int out_size, void* d_ws, size_t ws_size,
                              hipStream_t stream) {
    (void)in_sizes; (void)n_in; (void)out_size; (void)ws_size;
    const float* plocs  = (const float*)d_in[0];   // [B,P,4]
    const float* scores = (const float*)d_in[1];   // [B,P,C]
    const float* boxes  = (const float*)d_in[2];   // [B,M,4]
    const int*   labels = (const int*)d_in[3];     // [B,M]
    const float* priors = (const float*)d_in[4];   // [P,4]

    char* ws = (char*)d_ws;
    constexpr size_t szBP = (size_t)B_ * P_ * 4;                 // 1,117,696 B
    float*              ov    = (float*)(ws);
    int*                obj   = (int*)(ws + szBP);
    float*              ceneg = (float*)(ws + 2 * szBP);
    unsigned long long* best  = (unsigned long long*)(ws + 3 * szBP);
    int*                npos  = (int*)(ws + 3 * szBP + B_ * M_ * 8);
    float*              scal  = (float*)(ws + 3 * szBP + B_ * M_ * 8 + B_ * 4);
    float* out = (float*)d_out;

    const int pBlocks = (P_ + 255) / 256;          // 35

    k_init <<<1, 256, 0, stream>>>(best, npos, scal);
    k_match<<<dim3(pBlocks, B_), 256, 0, stream>>>((const float4*)priors,
                                                   (const float4*)boxes,
                                                   ov, obj, best);
    k_force<<<(B_ * M_ + 63) / 64, 64, 0, stream>>>(best, ov, obj);
    k_loc  <<<dim3(pBlocks, B_), 256, 0, stream>>>((const float4*)priors,
                                                   (const float4*)boxes, labels,
                                                   (const float4*)plocs,
                                                   ov, obj, scal, npos);
    k_ce   <<<dim3((P_ + 7) / 8, B_), 256, 0, stream>>>(scores, labels, ov, obj,
                                                        ceneg, scal);
    k_sort <<<B_, 1024, 0, stream>>>(ceneg, npos, scal);
    k_final<<<1, 32, 0, stream>>>(scal, npos, out);
}